// WRPLinear_40991167873568
// MI455X (gfx1250) — compile-verified
//
#include <hip/hip_runtime.h>

typedef float v2f __attribute__((ext_vector_type(2)));
typedef float v4f __attribute__((ext_vector_type(4)));
typedef float v8f __attribute__((ext_vector_type(8)));

#define D_IN  4096
#define D_OUT 4096

// Block tile: 128 (M) x 128 (N), K-tile 32, double-buffered LDS.
// 8 waves per block: 4 in M x 2 in N; each wave owns a 32x64 sub-tile
// = 2x4 grid of 16x16 f32 WMMA accumulators.
#define BM 128
#define BN 128
#define KT 32
#define LDA 36   // padded LDS row stride (floats): 16B-aligned rows,
                 // conflict-free for the 16x4 f32 fragment read pattern

#if __has_builtin(__builtin_amdgcn_global_load_async_to_lds_b128)
#define HAS_ASYNC_LDS 1
// Match the builtin's expected pointee type exactly (from the clang
// diagnostic): a GCC-style 16-byte int vector.
typedef int gv4i __attribute__((vector_size(16)));
#else
#define HAS_ASYNC_LDS 0
#endif

__device__ __forceinline__ void wait_async_zero() {
#if __has_builtin(__builtin_amdgcn_s_wait_asynccnt)
    __builtin_amdgcn_s_wait_asynccnt(0);
#else
    asm volatile("s_wait_asynccnt 0x0" ::: "memory");
#endif
}

__global__ __launch_bounds__(256)
void wrp_combine_w(const float* __restrict__ a, const float* __restrict__ b,
                   float* __restrict__ o) {
    size_t i = ((size_t)blockIdx.x * 256 + threadIdx.x) * 4;
    v4f va = *(const v4f*)(a + i);
    v4f vb = *(const v4f*)(b + i);
    *(v4f*)(o + i) = va + vb;
}

template <bool DUAL>
__global__ __launch_bounds__(256)
void wrp_gemm_f32(const float* __restrict__ X,
                  const float* __restrict__ W0,
                  const float* __restrict__ W1,
                  const float* __restrict__ bias,
                  float* __restrict__ Y) {
    __shared__ float lA[2][BM * LDA];
    __shared__ float lB[2][BN * LDA];

    const int tid   = threadIdx.x;
    const int wave  = tid >> 5;        // 0..7 (wave32)
    const int lane  = tid & 31;
    const int wm    = wave & 3;        // wave row: 4 waves in M
    const int wn    = wave >> 2;       // wave col: 2 waves in N
    const int l15   = lane & 15;
    const int lhalf = lane >> 4;       // 0/1: K-half for A/B fragments

    const int m_blk = blockIdx.y * BM;
    const int n_blk = blockIdx.x * BN;

    // Fill coordinates: each thread moves 4x b128 per tile per matrix.
    // chunk j covers rows frow + 32*j, columns fcol..fcol+3 of the K-tile.
    const int frow = tid >> 3;         // 0..31
    const int fcol = (tid & 7) << 2;   // 0,4,...,28
    const float* gA = X + (size_t)(m_blk + frow) * D_IN + fcol;
    const float* gW0 = W0 + (size_t)(n_blk + frow) * D_IN + fcol;
    const float* gW1 = DUAL ? (W1 + (size_t)(n_blk + frow) * D_IN + fcol)
                            : nullptr;
    const int lfill = frow * LDA + fcol;        // LDS word offset of chunk 0
    const size_t grow32 = (size_t)32 * D_IN;    // 32-row global stride
    const int lrow32 = 32 * LDA;                // 32-row LDS stride

    v8f acc[2][4];
#pragma unroll
    for (int mi = 0; mi < 2; ++mi)
#pragma unroll
        for (int ni = 0; ni < 4; ++ni)
            acc[mi][ni] = v8f{};

#if HAS_ASYNC_LDS
    // -------- async fill: global -> LDS via ASYNCcnt (no VGPR staging)
    auto stage_async = [&](int buf, int k0) {
#pragma unroll
        for (int j = 0; j < 4; ++j) {
            __builtin_amdgcn_global_load_async_to_lds_b128(
                (__attribute__((address_space(1))) gv4i*)(void*)(
                    gA + k0 + (size_t)j * grow32),
                (__attribute__((address_space(3))) gv4i*)(void*)(
                    &lA[buf][lfill + j * lrow32]),
                0, 0);
            __builtin_amdgcn_global_load_async_to_lds_b128(
                (__attribute__((address_space(1))) gv4i*)(void*)(
                    gW0 + k0 + (size_t)j * grow32),
                (__attribute__((address_space(3))) gv4i*)(void*)(
                    &lB[buf][lfill + j * lrow32]),
                0, 0);
        }
    };
#endif
    // -------- register-staged fill (used when DUAL, or no async builtin)
    v4f ra[4], rb[4];
    auto stage_load = [&](int k0) {
#pragma unroll
        for (int j = 0; j < 4; ++j) {
            ra[j] = __builtin_nontemporal_load(
                (const v4f*)(gA + k0 + (size_t)j * grow32));
            v4f w = *(const v4f*)(gW0 + k0 + (size_t)j * grow32);
            if (DUAL) w = w + *(const v4f*)(gW1 + k0 + (size_t)j * grow32);
            rb[j] = w;
        }
    };
    auto stage_store = [&](int buf) {
#pragma unroll
        for (int j = 0; j < 4; ++j) {
            *(v4f*)&lA[buf][lfill + j * lrow32] = ra[j];
            *(v4f*)&lB[buf][lfill + j * lrow32] = rb[j];
        }
    };

    auto compute = [&](int buf) {
#pragma unroll
        for (int kk = 0; kk < KT; kk += 4) {
            const int kc = kk + (lhalf << 1);
            v2f af[2];
            v2f bf[4];
#pragma unroll
            for (int mi = 0; mi < 2; ++mi)
                af[mi] = *(const v2f*)&lA[buf][(wm * 32 + mi * 16 + l15) * LDA + kc];
#pragma unroll
            for (int ni = 0; ni < 4; ++ni)
                bf[ni] = *(const v2f*)&lB[buf][(wn * 64 + ni * 16 + l15) * LDA + kc];
#pragma unroll
            for (int mi = 0; mi < 2; ++mi)
#pragma unroll
                for (int ni = 0; ni < 4; ++ni)
                    acc[mi][ni] = __builtin_amdgcn_wmma_f32_16x16x4_f32(
                        false, af[mi], false, bf[ni], (short)0, acc[mi][ni],
                        false, false);
        }
    };

    // -------- prologue: fill buffer 0 with K-tile 0
#if HAS_ASYNC_LDS
    if (!DUAL) {
        stage_async(0, 0);
        wait_async_zero();
    } else
#endif
    {
        stage_load(0);
        stage_store(0);
    }
    __syncthreads();

    // -------- main loop: prefetch next tile into buf^1 while computing buf
    int buf = 0;
    for (int k0 = 0; k0 < D_IN; k0 += KT) {
        const bool has_next = (k0 + KT) < D_IN;
#if HAS_ASYNC_LDS
        if (!DUAL) {
            if (has_next) stage_async(buf ^ 1, k0 + KT);
            compute(buf);
            if (has_next) {
                wait_async_zero();
                __syncthreads();
            }
        } else
#endif
        {
            if (has_next) stage_load(k0 + KT);
            compute(buf);
            if (has_next) {
                stage_store(buf ^ 1);
                __syncthreads();
            }
        }
        buf ^= 1;
    }

    // -------- epilogue: C layout is (M = r + 8*lhalf, N = l15) per VGPR r
#pragma unroll
    for (int ni = 0; ni < 4; ++ni) {
        const int n = n_blk + wn * 64 + ni * 16 + l15;
        const float bv = bias[n];
#pragma unroll
        for (int mi = 0; mi < 2; ++mi) {
#pragma unroll
            for (int r = 0; r < 8; ++r) {
                const int m = m_blk + wm * 32 + mi * 16 + (lhalf << 3) + r;
                __builtin_nontemporal_store(acc[mi][ni][r] + bv,
                                            Y + (size_t)m * D_OUT + n);
            }
        }
    }
}

extern "C" void kernel_launch(void* const* d_in, const int* in_sizes, int n_in,
                              void* d_out, int out_size, void* d_ws,
                              size_t ws_size, hipStream_t stream) {
    const float* x      = (const float*)d_in[0];
    const float* W_semi = (const float*)d_in[1];
    const float* W_rec  = (const float*)d_in[2];
    const float* bias   = (const float*)d_in[3];
    float* y            = (float*)d_out;

    const int M = in_sizes[0] / D_IN;  // 8192 = 2 * 4096
    dim3 grid(D_OUT / BN, M / BM);

    const size_t wbytes = (size_t)D_IN * D_OUT * sizeof(float);  // 64 MiB
    if (ws_size >= wbytes) {
        // Pre-combine weights once into workspace: halves GEMM weight traffic.
        float* Wc = (float*)d_ws;
        const int nvec4_blocks = (D_IN * D_OUT / 4) / 256;  // exact division
        wrp_combine_w<<<nvec4_blocks, 256, 0, stream>>>(W_semi, W_rec, Wc);
        wrp_gemm_f32<false><<<grid, 256, 0, stream>>>(x, Wc, nullptr, bias, y);
    } else {
        wrp_gemm_f32<true><<<grid, 256, 0, stream>>>(x, W_semi, W_rec, bias, y);
    }
}